// Cheb_gcn_11879879544526
// MI455X (gfx1250) — compile-verified
//
#include <hip/hip_runtime.h>
#include <math.h>

// Problem dims (fixed by reference)
#define B_    64
#define CIN   64
#define COUT  64
#define N_    512
#define T_    12
#define K_    3
#define D_    10
#define JT    (N_*T_)        // 6144 flattened spatial*time
#define EPSF  1e-5f
#define BCHUNK 32
#define NBC   (B_/BCHUNK)

typedef __attribute__((ext_vector_type(2))) float v2f;
typedef __attribute__((ext_vector_type(8))) float v8f;

// ---------------------------------------------------------------------------
// Stage 1: adaptive adjacency  adp = softmax(mask(relu(nv1 @ nv2)), axis=1)
// ---------------------------------------------------------------------------
__global__ void adp_kernel(const float* __restrict__ nv1,
                           const float* __restrict__ nv2,
                           float* __restrict__ adp) {
  __shared__ float red[256];
  const int n   = blockIdx.x;
  const int tid = threadIdx.x;

  float nv1row[D_];
#pragma unroll
  for (int d = 0; d < D_; ++d) nv1row[d] = nv1[n * D_ + d];

  float v[2];
#pragma unroll
  for (int i = 0; i < 2; ++i) {
    const int c = tid + i * 256;
    float s = 0.f;
#pragma unroll
    for (int d = 0; d < D_; ++d) s += nv1row[d] * nv2[d * N_ + c];
    s = fmaxf(s, 0.f);
    v[i] = (s == 0.f) ? -1e10f : s;   // mask exact zeros like the reference
  }

  float m = fmaxf(v[0], v[1]);
  red[tid] = m; __syncthreads();
  for (int o = 128; o > 0; o >>= 1) {
    if (tid < o) red[tid] = fmaxf(red[tid], red[tid + o]);
    __syncthreads();
  }
  m = red[0]; __syncthreads();

  float e[2];
  float s = 0.f;
#pragma unroll
  for (int i = 0; i < 2; ++i) { e[i] = expf(v[i] - m); s += e[i]; }
  red[tid] = s; __syncthreads();
  for (int o = 128; o > 0; o >>= 1) {
    if (tid < o) red[tid] += red[tid + o];
    __syncthreads();
  }
  const float inv = 1.f / red[0];
#pragma unroll
  for (int i = 0; i < 2; ++i) adp[(size_t)n * N_ + tid + i * 256] = e[i] * inv;
}

// ---------------------------------------------------------------------------
// Stage 2: Tk = softmax(alpha)[0]*cheb + [1]*adp, stored PAIR-PACKED:
//   tkp[k][n>>1][m][n&1]  so a lane's (K,K+1) B-pair is one contiguous b64.
// ---------------------------------------------------------------------------
__global__ void tk_kernel(const float* __restrict__ cheb,
                          const float* __restrict__ alpha,
                          const float* __restrict__ adp,
                          float* __restrict__ tkp) {
  const int idx = blockIdx.x * blockDim.x + threadIdx.x;   // < K_*N_*N_
  const int k  = idx / (N_ * N_);
  const int nm = idx % (N_ * N_);
  const int n  = nm / N_;
  const int m  = nm % N_;
  const float a0 = alpha[k * 2], a1 = alpha[k * 2 + 1];
  const float mx = fmaxf(a0, a1);
  const float e0 = expf(a0 - mx), e1 = expf(a1 - mx);
  const float inv = 1.f / (e0 + e1);
  const float val = (e0 * inv) * cheb[idx] + (e1 * inv) * adp[nm];
  tkp[(size_t)k * N_ * N_ + (size_t)(n >> 1) * (N_ * 2) + m * 2 + (n & 1)] = val;
}

// ---------------------------------------------------------------------------
// Stage 3a: channel mixing  y[bloc,o,t,n] = sum_c Theta_k[c,o]*relu(x[b,c,n,t])
// y is stored t-major/n-innermost so the spatial GEMM's A K-pairs are b64.
// One wave = 16(o) x 64(jt) strip; K=c loop by 4.
// ---------------------------------------------------------------------------
__global__ void chanmix_kernel(const float* __restrict__ x,      // [B,CIN,N,T]
                               const float* __restrict__ ThetaK, // [CIN,COUT]
                               float* __restrict__ y,            // [BCHUNK,COUT,T,N]
                               int bbase) {
  const int wv   = (blockIdx.x * blockDim.x + threadIdx.x) >> 5;
  const int lane = threadIdx.x & 31;
  const int half = lane >> 4;
  const int l16  = lane & 15;

  // waves: bloc(BCHUNK) x otile(4) x jstrip(96 of width 64)
  const int bloc = wv / (4 * 96);
  const int rem  = wv % (4 * 96);
  const int o0   = (rem / 96) * 16;
  const int j0   = (rem % 96) * 64;        // jt = t*512 + n
  const int t    = j0 >> 9;                // strip lies inside one t
  const int nb   = j0 & (N_ - 1);

  const float* xb = x + (size_t)(bbase + bloc) * CIN * JT;

  v8f acc[4] = {};
  for (int c0 = 0; c0 < CIN; c0 += 4) {
    const int ca = c0 + 2 * half;          // ISA A-layout: lanes16-31 hold K+2
    v2f a;
    a.x = ThetaK[(size_t)ca * COUT + (o0 + l16)];
    a.y = ThetaK[(size_t)(ca + 1) * COUT + (o0 + l16)];
#pragma unroll
    for (int f = 0; f < 4; ++f) {
      const int n = nb + f * 16 + l16;
      v2f b;
      b.x = fmaxf(xb[(size_t)ca * JT + n * T_ + t], 0.f);        // fused relu
      b.y = fmaxf(xb[(size_t)(ca + 1) * JT + n * T_ + t], 0.f);
      acc[f] = __builtin_amdgcn_wmma_f32_16x16x4_f32(
          false, a, false, b, (short)0, acc[f], false, false);
    }
  }

  float* yb = y + (size_t)bloc * COUT * JT;
#pragma unroll
  for (int f = 0; f < 4; ++f)
#pragma unroll
    for (int v = 0; v < 8; ++v)
      yb[(size_t)(o0 + v + 8 * half) * JT + (j0 + f * 16 + l16)] = acc[f][v];
}

// ---------------------------------------------------------------------------
// Stage 3b: spatial mixing  out[b,o,m,t] (+)= sum_n Y[(o,t),n] * Tk[n,m]
// GEMM M=(o,t)=768, K=n=512, N=m=512 per b.  Block tile 128x256, 8 waves
// (2 row x 4 col), wave tile 64x64 -> 16 WMMA per K-step vs 8 b64 loads.
// ---------------------------------------------------------------------------
__global__ void spatial_kernel(const float* __restrict__ y,    // [BCHUNK,COUT,T,N]
                               const float* __restrict__ tkp,  // packed k slice
                               float* __restrict__ out,        // [B,COUT,N,T]
                               int bbase, int accumulate) {
  const int blk   = blockIdx.x;            // BCHUNK * rtile(6) * mtile(2)
  const int bloc  = blk / 12;
  const int rtile = (blk % 12) / 2;
  const int mtile = blk % 2;
  const int wv    = threadIdx.x >> 5;
  const int lane  = threadIdx.x & 31;
  const int half  = lane >> 4;
  const int l16   = lane & 15;

  const int r0 = rtile * 128 + (wv >> 2) * 64;   // rows = (o,t) flat, o-major
  const int m0 = mtile * 256 + (wv & 3) * 64;

  const float* yb = y + (size_t)bloc * COUT * JT;

  int rowoff[4];                                  // row r -> o*JT + t*N_
#pragma unroll
  for (int fr = 0; fr < 4; ++fr) {
    const int r = r0 + fr * 16 + l16;
    rowoff[fr] = (r / T_) * JT + (r % T_) * N_;
  }
  size_t coloff[4];                               // packed Tk column offsets
#pragma unroll
  for (int fc = 0; fc < 4; ++fc)
    coloff[fc] = (size_t)(m0 + fc * 16 + l16) * 2;

  v8f acc[4][4] = {};
  for (int n0 = 0; n0 < N_; n0 += 4) {
    const int na = n0 + 2 * half;                 // even
    v2f a[4], b[4];
#pragma unroll
    for (int fr = 0; fr < 4; ++fr)                // contiguous K-pair: b64
      a[fr] = *(const v2f*)(yb + rowoff[fr] + na);
    const float* tkrow = tkp + (size_t)(na >> 1) * (N_ * 2);
#pragma unroll
    for (int fc = 0; fc < 4; ++fc)                // contiguous K-pair: b64
      b[fc] = *(const v2f*)(tkrow + coloff[fc]);
#pragma unroll
    for (int fr = 0; fr < 4; ++fr)
#pragma unroll
      for (int fc = 0; fc < 4; ++fc)
        acc[fr][fc] = __builtin_amdgcn_wmma_f32_16x16x4_f32(
            false, a[fr], false, b[fc], (short)0, acc[fr][fc], false, false);
  }

  const int bglob = bbase + bloc;
#pragma unroll
  for (int fr = 0; fr < 4; ++fr) {
#pragma unroll
    for (int v = 0; v < 8; ++v) {
      const int r = r0 + fr * 16 + v + 8 * half;
      const int o = r / T_, t = r % T_;
      const size_t base = (((size_t)bglob * COUT + o) * N_) * T_ + t;
#pragma unroll
      for (int fc = 0; fc < 4; ++fc) {
        const size_t addr = base + (size_t)(m0 + fc * 16 + l16) * T_;
        const float val = acc[fr][fc][v];
        out[addr] = accumulate ? (out[addr] + val) : val;
      }
    }
  }
}

// ---------------------------------------------------------------------------
// Stage 4: deterministic batchnorm (stats over relu(out) per channel o)
// ---------------------------------------------------------------------------
__global__ void stats1_kernel(const float* __restrict__ out,
                              float* __restrict__ psum, float* __restrict__ psq) {
  const int o = blockIdx.x >> 6;   // blockIdx = o*64 + b
  const int b = blockIdx.x & 63;
  const float* p = out + (((size_t)b * COUT + o) * N_) * T_;   // 6144 contiguous
  float s = 0.f, q = 0.f;
  for (int i = threadIdx.x; i < JT; i += 256) {
    const float v = fmaxf(p[i], 0.f);
    s += v; q += v * v;
  }
  __shared__ float rs[256], rq[256];
  rs[threadIdx.x] = s; rq[threadIdx.x] = q; __syncthreads();
  for (int ofs = 128; ofs > 0; ofs >>= 1) {
    if (threadIdx.x < ofs) {
      rs[threadIdx.x] += rs[threadIdx.x + ofs];
      rq[threadIdx.x] += rq[threadIdx.x + ofs];
    }
    __syncthreads();
  }
  if (threadIdx.x == 0) { psum[blockIdx.x] = rs[0]; psq[blockIdx.x] = rq[0]; }
}

__global__ void stats2_kernel(const float* __restrict__ psum,
                              const float* __restrict__ psq,
                              float* __restrict__ meanistd) {
  const int o = blockIdx.x;
  const int t = threadIdx.x;     // 64 threads
  __shared__ float rs[64], rq[64];
  rs[t] = psum[o * 64 + t]; rq[t] = psq[o * 64 + t]; __syncthreads();
  for (int ofs = 32; ofs > 0; ofs >>= 1) {
    if (t < ofs) { rs[t] += rs[t + ofs]; rq[t] += rq[t + ofs]; }
    __syncthreads();
  }
  if (t == 0) {
    const float cnt = (float)B_ * (float)N_ * (float)T_;
    const float mean = rs[0] / cnt;
    const float var  = rq[0] / cnt - mean * mean;   // biased, matches jnp.var
    meanistd[o]      = mean;
    meanistd[64 + o] = rsqrtf(var + EPSF);
  }
}

__global__ void bnorm_kernel(float* __restrict__ out,
                             const float* __restrict__ meanistd) {
  const size_t i = (size_t)blockIdx.x * 256 + threadIdx.x;
  const int o = (int)((i / (size_t)(N_ * T_)) & 63);   // COUT == 64
  const float v = fmaxf(out[i], 0.f);                  // relu fused here
  out[i] = (v - meanistd[o]) * meanistd[64 + o];
}

// ---------------------------------------------------------------------------
extern "C" void kernel_launch(void* const* d_in, const int* in_sizes, int n_in,
                              void* d_out, int out_size, void* d_ws, size_t ws_size,
                              hipStream_t stream) {
  const float* x     = (const float*)d_in[0];
  const float* cheb  = (const float*)d_in[1];
  const float* nv1   = (const float*)d_in[2];
  const float* nv2   = (const float*)d_in[3];
  const float* alpha = (const float*)d_in[4];
  const float* Theta = (const float*)d_in[5];
  float* out = (float*)d_out;

  // Workspace layout (floats): ~55 MB total
  float* ws   = (float*)d_ws;
  float* adp  = ws;                                   // N*N        = 262144
  float* tkp  = adp + (size_t)N_ * N_;                // K*N*N      = 786432
  float* y    = tkp + (size_t)K_ * N_ * N_;           // 32*64*6144 = 12582912
  float* psum = y + (size_t)BCHUNK * COUT * JT;       // 4096
  float* psq  = psum + (size_t)COUT * B_;             // 4096
  float* mi   = psq + (size_t)COUT * B_;              // 128

  adp_kernel<<<N_, 256, 0, stream>>>(nv1, nv2, adp);
  tk_kernel<<<(K_ * N_ * N_) / 256, 256, 0, stream>>>(cheb, alpha, adp, tkp);

  for (int bc = 0; bc < NBC; ++bc) {
    const int bbase = bc * BCHUNK;
    for (int k = 0; k < K_; ++k) {
      // BCHUNK * 4 o-tiles * 96 j-strips waves / 8 waves per block
      chanmix_kernel<<<BCHUNK * 4 * 96 / 8, 256, 0, stream>>>(
          x, Theta + (size_t)k * CIN * COUT, y, bbase);
      // BCHUNK * 6 r-tiles * 2 m-tiles blocks
      spatial_kernel<<<BCHUNK * 6 * 2, 256, 0, stream>>>(
          y, tkp + (size_t)k * N_ * N_, out, bbase, (k > 0) ? 1 : 0);
    }
  }

  stats1_kernel<<<COUT * B_, 256, 0, stream>>>(out, psum, psq);
  stats2_kernel<<<COUT, 64, 0, stream>>>(psum, psq, mi);
  bnorm_kernel<<<(int)(((size_t)B_ * COUT * N_ * T_) / 256), 256, 0, stream>>>(out, mi);
}